// DoubleFeatureTransformerSlice_33492154974541
// MI455X (gfx1250) — compile-verified
//
#include <hip/hip_runtime.h>
#include <stdint.h>

// ---------------------------------------------------------------------------
// DoubleFeatureTransformerSlice for MI455X (gfx1250)
//
// out[s][b][d] = bias[d] + sum_k val[s][b][k] * W[idx[s][b][k]][d]
//   BATCH=8192, K=32, D=1024, NUM_INPUTS=22528, two slices.
//
// L2-resident gather (88 MiB table in 192 MB L2); 2 GiB of row gathers are
// moved by the CDNA5 Tensor Data Mover in GATHER MODE (8 rows / descriptor,
// 32-bit indices) into LDS, overlapped with FMA accumulation via TENSORcnt
// software pipelining. Indices AND per-feature values are scalar-loaded
// (readfirstlane -> s_load_b512) so the VALU loop is pure
// ds_load_b128 + v_dual_fmac with SGPR scale operands. Outputs stored
// non-temporally to preserve the weight table's L2 residency.
// ---------------------------------------------------------------------------

#define NUM_INPUTS   22528
#define NUM_OUTPUTS  1024
#define BATCH        8192
#define K_ACTIVE     32

#define WAVES_PER_BLOCK 4
#define D_PER_WAVE      256                         // quarter of D=1024
#define ROWS_PER_DESC   8                           // 32-bit index gather mode
#define DESC_PER_WAVE   4                           // 4*8 = 32 rows
#define LDS_DESC_BYTES  (ROWS_PER_DESC * D_PER_WAVE * 4)   // 8 KB
#define LDS_WAVE_BYTES  (K_ACTIVE * D_PER_WAVE * 4)        // 32 KB
#define LDS_BLOCK_BYTES (WAVES_PER_BLOCK * LDS_WAVE_BYTES) // 128 KB

typedef __attribute__((ext_vector_type(4))) unsigned int v4u;
typedef __attribute__((ext_vector_type(8))) int          v8i;
typedef __attribute__((ext_vector_type(4))) int          v4i;
typedef __attribute__((ext_vector_type(4))) float        v4f;

__device__ __forceinline__ int rfl(int x) {
    return __builtin_amdgcn_readfirstlane(x);
}

extern "C" __global__ void __launch_bounds__(128)
nnue_ft_tdm_kernel(const int*   __restrict__ idx0,
                   const float* __restrict__ val0,
                   const int*   __restrict__ idx1,
                   const float* __restrict__ val1,
                   const float* __restrict__ weight,
                   const float* __restrict__ bias,
                   float*       __restrict__ out)
{
    extern __shared__ float smem[];   // only LDS object -> dynamic segment at offset 0

    const unsigned lane      = threadIdx.x & 31u;
    const unsigned waveInBlk = (unsigned)rfl((int)(threadIdx.x >> 5));
    const unsigned gw        = blockIdx.x * WAVES_PER_BLOCK + waveInBlk; // 0..65535
    const unsigned quarter   = gw & 3u;            // which 256-column slab of D
    const unsigned bs        = gw >> 2;            // 0..16383
    const unsigned b         = bs & (BATCH - 1u);
    const unsigned slice     = bs >> 13;           // 0 or 1 (uniform per wave)

    const int*   idxp = (slice == 0 ? idx0 : idx1) + (size_t)b * K_ACTIVE;
    const float* valp = (slice == 0 ? val0 : val1) + (size_t)b * K_ACTIVE;

    // Scalar-load all 32 indices and values (wave-uniform addresses ->
    // s_load_b512 pairs; rfl pins results to SGPRs).
    int   sidx[K_ACTIVE];
    float sval[K_ACTIVE];
#pragma unroll
    for (int k = 0; k < K_ACTIVE; ++k) {
        sidx[k] = rfl(idxp[k]);
        sval[k] = __uint_as_float((unsigned)rfl((int)__float_as_uint(valp[k])));
    }

    // ---------------- build + issue 4 TDM gather descriptors ----------------
    const unsigned ldsBase = waveInBlk * (unsigned)LDS_WAVE_BYTES;
    const uint64_t gaddr   = (uint64_t)(uintptr_t)weight
                           + (uint64_t)quarter * (D_PER_WAVE * 4u); // column slab start

    // D# group 1 (256b): data_size=4B; tensor_dim0=1024; tensor_dim1=22528;
    // tile_dim0=256; tile_dim1=#indices=8; dim0 stride=1024 elements.
    v8i g1;
    g1[0] = 0x00020000;                                   // data_size=2 (4 bytes)
    g1[1] = (int)((NUM_OUTPUTS & 0xFFFF) << 16);          // tensor_dim0[15:0]
    g1[2] = (int)(((NUM_OUTPUTS >> 16) & 0xFFFF) |
                  ((NUM_INPUTS & 0xFFFF) << 16));         // dim0 hi | tensor_dim1 lo
    g1[3] = (int)(((NUM_INPUTS >> 16) & 0xFFFF) |
                  ((unsigned)D_PER_WAVE << 16));          // dim1 hi | tile_dim0
    g1[4] = ROWS_PER_DESC;                                // tile_dim1 (#valid indices)
    g1[5] = NUM_OUTPUTS;                                  // tensor_dim0_stride lo
    g1[6] = 0;
    g1[7] = 0;

    // trailing 256-bit group (6-arg builtin form); unused for 2-D gather
    v8i gtail;
    gtail[0] = 0; gtail[1] = 0; gtail[2] = 0; gtail[3] = 0;
    gtail[4] = 0; gtail[5] = 0; gtail[6] = 0; gtail[7] = 0;

#pragma unroll
    for (int g = 0; g < DESC_PER_WAVE; ++g) {
        // D# group 0: count=1, gather_mode=1 (bit31), 32-bit indices (bit30),
        // lds_addr, global_addr, type=2.
        v4u g0;
        g0[0] = 0xC0000001u;
        g0[1] = ldsBase + (unsigned)g * LDS_DESC_BYTES;
        g0[2] = (unsigned)(gaddr & 0xFFFFFFFFu);
        g0[3] = (unsigned)((gaddr >> 32) & 0x01FFFFFFu) | 0x80000000u;

        // D# groups 2/3: eight 32-bit row indices
        v4i g2, g3;
        g2[0] = sidx[g * 8 + 0];
        g2[1] = sidx[g * 8 + 1];
        g2[2] = sidx[g * 8 + 2];
        g2[3] = sidx[g * 8 + 3];
        g3[0] = sidx[g * 8 + 4];
        g3[1] = sidx[g * 8 + 5];
        g3[2] = sidx[g * 8 + 6];
        g3[3] = sidx[g * 8 + 7];

        __builtin_amdgcn_tensor_load_to_lds(g0, g1, g2, g3, gtail, /*cpol=*/0);
    }

    // ---------------- accumulators seeded with bias ----------------
    const float* bp = bias + quarter * D_PER_WAVE;
    v4f acc0 = *(const v4f*)(bp + lane * 4);        // cols [0,128) of slab
    v4f acc1 = *(const v4f*)(bp + 128 + lane * 4);  // cols [128,256) of slab

    const float* wbuf = smem + waveInBlk * (LDS_WAVE_BYTES / 4);

    // ---------------- pipelined consume: wait desc g, FMA its 8 rows --------
#define STAGE(G, WAITSTR)                                                     \
    {                                                                         \
        asm volatile("s_wait_tensorcnt " WAITSTR ::: "memory");               \
        const float* rowsG = wbuf + (G) * (LDS_DESC_BYTES / 4);               \
        _Pragma("unroll")                                                     \
        for (int r = 0; r < ROWS_PER_DESC; ++r) {                             \
            float vk = sval[(G) * 8 + r];                                     \
            v4f w0 = *(const v4f*)(rowsG + r * D_PER_WAVE + lane * 4);        \
            v4f w1 = *(const v4f*)(rowsG + r * D_PER_WAVE + 128 + lane * 4);  \
            acc0 += vk * w0;                                                  \
            acc1 += vk * w1;                                                  \
        }                                                                     \
    }

    STAGE(0, "3")
    STAGE(1, "2")
    STAGE(2, "1")
    STAGE(3, "0")
#undef STAGE

    // ---------------- write out (non-temporal: keep weights in L2) ----------
    float* op = out + (size_t)slice * BATCH * NUM_OUTPUTS
                    + (size_t)b * NUM_OUTPUTS
                    + quarter * D_PER_WAVE;
    __builtin_nontemporal_store(acc0, (v4f*)(op + lane * 4));
    __builtin_nontemporal_store(acc1, (v4f*)(op + 128 + lane * 4));
}

extern "C" void kernel_launch(void* const* d_in, const int* in_sizes, int n_in,
                              void* d_out, int out_size, void* d_ws, size_t ws_size,
                              hipStream_t stream)
{
    const int*   idx0   = (const int*)  d_in[0];
    const float* val0   = (const float*)d_in[1];
    const int*   idx1   = (const int*)  d_in[2];
    const float* val1   = (const float*)d_in[3];
    const float* weight = (const float*)d_in[4];
    const float* bias   = (const float*)d_in[5];
    float*       out    = (float*)d_out;

    // 2 slices * 8192 batch * 4 quarter-slabs = 65536 waves; 4 waves/block
    dim3 grid(16384), block(128);
    hipLaunchKernelGGL(nnue_ft_tdm_kernel, grid, block, LDS_BLOCK_BYTES, stream,
                       idx0, val0, idx1, val1, weight, bias, out);
}